// LocalGNN_20727512170561
// MI455X (gfx1250) — compile-verified
//
#include <hip/hip_runtime.h>
#include <hip/hip_bf16.h>

typedef float v2f __attribute__((ext_vector_type(2)));
typedef float v8f __attribute__((ext_vector_type(8)));

#define HID 128
#define LDS_STRIDE 132   // padded row stride (floats) to avoid LDS bank conflicts

// ---------------- utility kernels ----------------

__global__ void zero_i32_kernel(int* __restrict__ p, int n) {
  int i = blockIdx.x * blockDim.x + threadIdx.x;
  if (i < n) p[i] = 0;
}

__global__ void count_deg_kernel(const int* __restrict__ src, const int* __restrict__ dst,
                                 int* __restrict__ cs, int* __restrict__ cd, int E) {
  int e = blockIdx.x * blockDim.x + threadIdx.x;
  if (e < E) {
    atomicAdd(&cs[src[e]], 1);
    atomicAdd(&cd[dst[e]], 1);
  }
}

__global__ void norm_kernel(const int* __restrict__ cs, const int* __restrict__ cd,
                            float* __restrict__ ns, float* __restrict__ nd, int n) {
  int i = blockIdx.x * blockDim.x + threadIdx.x;
  if (i < n) {
    int a = cs[i]; if (a < 1) a = 1;
    int b = cd[i]; if (b < 1) b = 1;
    ns[i] = rsqrtf((float)a);
    nd[i] = rsqrtf((float)b);
  }
}

// Single-workgroup exclusive scan over n counts -> offsets[0..n]
__global__ void scan_kernel(const int* __restrict__ counts, int* __restrict__ offsets, int n) {
  __shared__ int sm[1024];
  __shared__ int carry_s;
  int tid = threadIdx.x;
  if (tid == 0) carry_s = 0;
  __syncthreads();
  for (int base = 0; base < n; base += 1024) {
    int v = (base + tid < n) ? counts[base + tid] : 0;
    sm[tid] = v;
    __syncthreads();
    for (int off = 1; off < 1024; off <<= 1) {
      int t = (tid >= off) ? sm[tid - off] : 0;
      __syncthreads();
      sm[tid] += t;
      __syncthreads();
    }
    int incl  = sm[tid];
    int carry = carry_s;
    if (base + tid < n) offsets[base + tid] = carry + incl - v;  // exclusive
    __syncthreads();
    if (tid == 1023) carry_s = carry + incl;  // incl of tid 1023 == chunk total
    __syncthreads();
  }
  if (tid == 0) offsets[n] = carry_s;
}

__global__ void fill_csr_kernel(const int* __restrict__ src, const int* __restrict__ dst,
                                const int* __restrict__ rowoff, int* __restrict__ cursor,
                                int* __restrict__ colidx, int E) {
  int e = blockIdx.x * blockDim.x + threadIdx.x;
  if (e < E) {
    int d = dst[e];
    int p = atomicAdd(&cursor[d], 1);
    colidx[rowoff[d] + p] = src[e];
  }
}

// ---------------- SpMM: agg[d] = norm_dst[d] * sum_{s in N(d)} norm_src[s] * h[s] ----------------
// One wave per destination row; each lane owns a float4 slice of the 128-dim feature.

__global__ __launch_bounds__(256) void spmm_kernel(
    const float* __restrict__ hin, const int* __restrict__ rowoff,
    const int* __restrict__ colidx, const float* __restrict__ nsrc,
    const float* __restrict__ ndst, float* __restrict__ agg, int n) {
  int gwave  = (blockIdx.x * blockDim.x + threadIdx.x) >> 5;
  int lane   = threadIdx.x & 31;
  int nwaves = (gridDim.x * blockDim.x) >> 5;
  for (int row = gwave; row < n; row += nwaves) {
    int beg = rowoff[row], end = rowoff[row + 1];
    float ax = 0.f, ay = 0.f, az = 0.f, aw = 0.f;
    for (int j = beg; j < end; ++j) {
      int s = colidx[j];
      if (j + 1 < end) {  // prefetch next neighbor row (global_prefetch_b8)
        int s2 = colidx[j + 1];
        __builtin_prefetch(hin + (size_t)s2 * HID + lane * 4, 0, 0);
      }
      float w = nsrc[s];
      const float4 v = *(const float4*)(hin + (size_t)s * HID + lane * 4);
      ax = fmaf(w, v.x, ax);
      ay = fmaf(w, v.y, ay);
      az = fmaf(w, v.z, az);
      aw = fmaf(w, v.w, aw);
    }
    float nd = ndst[row];
    float4 r = make_float4(ax * nd, ay * nd, az * nd, aw * nd);
    *(float4*)(agg + (size_t)row * HID + lane * 4) = r;
  }
}

// ---------------- GEMM + bias + ReLU:  out[M,128] = relu(A[M,K] @ W[K,128] + b) ----------------
// fp32 WMMA 16x16x4. Workgroup = 256 threads = 8 waves; each wave computes a 16x128 strip.
// W (and bias) staged in LDS with padded stride to kill bank conflicts.

__global__ __launch_bounds__(256) void gemm_bias_relu_kernel(
    const float* __restrict__ A, const float* __restrict__ W,
    const float* __restrict__ bias, float* __restrict__ out, int nrows, int K) {
  __shared__ float ldsW[HID * LDS_STRIDE];
  __shared__ float ldsB[HID];

  for (int idx = threadIdx.x; idx < K * HID; idx += 256) {
    int r = idx >> 7, c = idx & 127;
    ldsW[r * LDS_STRIDE + c] = W[idx];
  }
  if (threadIdx.x < HID) ldsB[threadIdx.x] = bias[threadIdx.x];
  __syncthreads();

  int waveId = threadIdx.x >> 5;
  int lane   = threadIdx.x & 31;
  int m      = lane & 15;   // M index within tile (A), N index within tile (B/C)
  int kh     = lane >> 4;   // K half: lanes 0-15 hold K{0,1}, lanes 16-31 hold K{2,3}
  int row0   = blockIdx.x * 128 + waveId * 16;
  if (row0 >= nrows) return;  // whole-wave uniform: EXEC stays all-ones for WMMA

  v8f acc[8];
#pragma unroll
  for (int t = 0; t < 8; ++t)
#pragma unroll
    for (int i = 0; i < 8; ++i) acc[t][i] = 0.f;

  const float* Arow = A + (size_t)(row0 + m) * K;
  for (int kb = 0; kb < K; kb += 4) {
    int kk = kb + 2 * kh;
    v2f a = *(const v2f*)(Arow + kk);  // A[m][kk], A[m][kk+1]
#pragma unroll
    for (int t = 0; t < 8; ++t) {
      v2f b;
      b.x = ldsW[kk * LDS_STRIDE + t * 16 + m];        // W[kk][n]
      b.y = ldsW[(kk + 1) * LDS_STRIDE + t * 16 + m];  // W[kk+1][n]
      acc[t] = __builtin_amdgcn_wmma_f32_16x16x4_f32(
          false, a, false, b, (short)0, acc[t], false, false);
    }
  }

  // C/D layout: VGPR i -> M = i (lanes 0-15) / i+8 (lanes 16-31); N = lane&15
  int rAdd = kh * 8;
#pragma unroll
  for (int t = 0; t < 8; ++t) {
    int col = t * 16 + m;
    float bv = ldsB[col];
#pragma unroll
    for (int i = 0; i < 8; ++i) {
      int row = row0 + i + rAdd;
      float v = acc[t][i] + bv;
      out[(size_t)row * HID + col] = fmaxf(v, 0.f);
    }
  }
}

// ---------------- launch ----------------

extern "C" void kernel_launch(void* const* d_in, const int* in_sizes, int n_in,
                              void* d_out, int out_size, void* d_ws, size_t ws_size,
                              hipStream_t stream) {
  const float* x_raw = (const float*)d_in[0];
  const int*   src   = (const int*)d_in[1];
  const int*   dst   = (const int*)d_in[2];
  const float* Wp    = (const float*)d_in[3];
  const float* bp    = (const float*)d_in[4];
  const float* Wl    = (const float*)d_in[5];
  const float* bl    = (const float*)d_in[6];
  float* out = (float*)d_out;

  const int IN = 64;
  const int N  = in_sizes[0] / IN;
  const int E  = in_sizes[1];
  const int L  = in_sizes[5] / (HID * HID);

  // workspace carve-up
  char* base = (char*)d_ws;
  size_t off = 0;
  auto carve = [&](size_t bytes) { void* p = base + off; off = (off + bytes + 511) & ~(size_t)511; return p; };
  int*   cnt_src = (int*)carve((size_t)N * 4);
  int*   cnt_dst = (int*)carve((size_t)N * 4);
  int*   cursor  = (int*)carve((size_t)N * 4);
  int*   rowoff  = (int*)carve((size_t)(N + 1) * 4);
  int*   colidx  = (int*)carve((size_t)E * 4);
  float* nsrc    = (float*)carve((size_t)N * 4);
  float* ndst    = (float*)carve((size_t)N * 4);
  float* hbuf    = (float*)carve((size_t)N * HID * 4);
  float* agg     = (float*)carve((size_t)N * HID * 4);
  (void)ws_size; (void)n_in; (void)out_size;

  const int TB = 256;
  int gN  = (N + TB - 1) / TB;
  int gE  = (E + TB - 1) / TB;
  int g3N = (3 * N + TB - 1) / TB;

  // cnt_src, cnt_dst, cursor are contiguous in carve order except for padding;
  // zero each individually to be safe about alignment padding.
  zero_i32_kernel<<<gN, TB, 0, stream>>>(cnt_src, N);
  zero_i32_kernel<<<gN, TB, 0, stream>>>(cnt_dst, N);
  zero_i32_kernel<<<gN, TB, 0, stream>>>(cursor, N);
  (void)g3N;

  count_deg_kernel<<<gE, TB, 0, stream>>>(src, dst, cnt_src, cnt_dst, E);
  norm_kernel<<<gN, TB, 0, stream>>>(cnt_src, cnt_dst, nsrc, ndst, N);
  scan_kernel<<<1, 1024, 0, stream>>>(cnt_dst, rowoff, N);
  fill_csr_kernel<<<gE, TB, 0, stream>>>(src, dst, rowoff, cursor, colidx, E);

  // projection: h = relu(x @ Wp + bp)
  int gGemm = (N + 127) / 128;
  gemm_bias_relu_kernel<<<gGemm, TB, 0, stream>>>(x_raw, Wp, bp, hbuf, N, IN);

  // SpMM waves: one wave per row
  int gSpmm = (N * 32 + TB - 1) / TB;

  for (int l = 0; l < L; ++l) {
    spmm_kernel<<<gSpmm, TB, 0, stream>>>(hbuf, rowoff, colidx, nsrc, ndst, agg, N);
    float* dstbuf = (l == L - 1) ? out : hbuf;
    gemm_bias_relu_kernel<<<gGemm, TB, 0, stream>>>(
        agg, Wl + (size_t)l * HID * HID, bl + (size_t)l * HID, dstbuf, N, HID);
  }
}